// GatedGATConv_37391985278999
// MI455X (gfx1250) — compile-verified
//
#include <hip/hip_runtime.h>
#include <hip/hip_bf16.h>

constexpr int kD  = 800;
constexpr int kNA = 20000;
constexpr int kNB = 24000;
constexpr int kNG = 256;

typedef __attribute__((ext_vector_type(2))) float v2f;
typedef __attribute__((ext_vector_type(8))) float v8f;

// =====================================================================
// WMMA f32 GEMM: per wave a 32x80 tile of Y[M,800] = X[M,800]@W[800,800]+b
// (2 M-subtiles x 5 N-subtiles = 10 accumulators; each B fragment feeds
//  2 WMMAs -> ~1.2 load-issues per WMMA instead of 2.2)
// MODE 0: Y = acc + b
// MODE 1: Y = beta*Y + alpha*(acc + b)
// MODE 2: segout[seg[row],col] += (acc+b) * invc[seg[row]]   (segment mean)
// =====================================================================
template <int MODE>
__global__ __launch_bounds__(128) void gemm800(
    const float* __restrict__ X, const float* __restrict__ W,
    const float* __restrict__ bias, float* __restrict__ Y, int M,
    float alpha, float beta, const int* __restrict__ seg,
    const float* __restrict__ invc, float* __restrict__ segout) {
  const int lane = threadIdx.x & 31;
  const int wid  = (blockIdx.x * blockDim.x + threadIdx.x) >> 5;
  const int mtiles = M >> 5;              // 32-row tiles
  if (wid >= mtiles * 10) return;
  const int mt = wid / 10;
  const int nt = wid - mt * 10;
  const int m0 = mt << 5;
  const int n0 = nt * 80;
  const int r   = lane & 15;   // row (A) / col (B) within 16
  const int hi  = lane >> 4;   // lane half selects K pair {0,1} vs {2,3}
  const int dk0 = hi << 1;

  v8f acc[2][5] = {};
  const float* xr0 = X + (size_t)(m0 + r) * kD + dk0;       // 8B aligned
  const float* xr1 = X + (size_t)(m0 + 16 + r) * kD + dk0;
  const float* wb  = W + (size_t)dk0 * kD + n0 + r;
  for (int k = 0; k < kD; k += 4) {
    const v2f a0 = *(const v2f*)(xr0 + k);   // A frag rows m0..m0+15
    const v2f a1 = *(const v2f*)(xr1 + k);   // A frag rows m0+16..m0+31
    const float* wp = wb + (size_t)k * kD;
#pragma unroll
    for (int j = 0; j < 5; ++j) {
      v2f b;
      b.x = wp[j * 16];        // W[k+dk0  ][n0+j*16+r]
      b.y = wp[j * 16 + kD];   // W[k+dk0+1][n0+j*16+r]
      acc[0][j] = __builtin_amdgcn_wmma_f32_16x16x4_f32(
          false, a0, false, b, (short)0, acc[0][j], false, false);
      acc[1][j] = __builtin_amdgcn_wmma_f32_16x16x4_f32(
          false, a1, false, b, (short)0, acc[1][j], false, false);
    }
  }
#pragma unroll
  for (int mi = 0; mi < 2; ++mi) {
#pragma unroll
    for (int j = 0; j < 5; ++j) {
      const int col = n0 + j * 16 + r;
      const float bv = bias[col];
#pragma unroll
      for (int rr = 0; rr < 8; ++rr) {
        const int row = m0 + mi * 16 + rr + (hi << 3);  // C/D: lanes>=16 -> +8
        const float v = acc[mi][j][rr] + bv;
        if (MODE == 0) {
          Y[(size_t)row * kD + col] = v;
        } else if (MODE == 1) {
          float* p = Y + (size_t)row * kD + col;
          *p = beta * (*p) + alpha * v;
        } else {
          const int g = seg[row];
          unsafeAtomicAdd(segout + (size_t)g * kD + col, v * invc[g]);
        }
      }
    }
  }
}

// ============================ elementwise helpers ====================
__global__ void zero_f32(float* p, size_t n) {
  size_t i = (size_t)blockIdx.x * blockDim.x + threadIdx.x;
  size_t s = (size_t)gridDim.x * blockDim.x;
  for (; i < n; i += s) p[i] = 0.0f;
}
__global__ void zero_i32(int* p, size_t n) {
  size_t i = (size_t)blockIdx.x * blockDim.x + threadIdx.x;
  size_t s = (size_t)gridDim.x * blockDim.x;
  for (; i < n; i += s) p[i] = 0;
}

// e_out[b,:] += Ah[i0,:] + Ah[i1,:] + Cu[bond_graph[b],:]
__global__ void bond_gather(float* __restrict__ e_out,
                            const float* __restrict__ Ah,
                            const float* __restrict__ Cu,
                            const int* __restrict__ bond_atoms,
                            const int* __restrict__ bond_graph) {
  const int b  = blockIdx.x;
  const int i0 = bond_atoms[2 * b];
  const int i1 = bond_atoms[2 * b + 1];
  const int g  = bond_graph[b];
  const float* a0 = Ah + (size_t)i0 * kD;
  const float* a1 = Ah + (size_t)i1 * kD;
  const float* cu = Cu + (size_t)g  * kD;
  float* out = e_out + (size_t)b * kD;
  for (int c = threadIdx.x; c < kD; c += blockDim.x)
    out[c] += a0[c] + a1[c] + cu[c];
}

// gated mailbox aggregation:
// h_out[a] = Dh[a] + sum_k sig(e_n[bk])*Eh[other_k] / (sum_k sig + eps) + Fu[g]
__global__ void atom_update(float* __restrict__ h_out,
                            const float* __restrict__ Dh,
                            const float* __restrict__ Eh,
                            const float* __restrict__ Fu,
                            const float* __restrict__ e_n,
                            const int* __restrict__ atom_bonds,
                            const int* __restrict__ bond_atoms,
                            const int* __restrict__ atom_graph) {
  const int a = blockIdx.x;
  const float* eb[4];
  const float* eo[4];
#pragma unroll
  for (int k = 0; k < 4; ++k) {
    const int bk = atom_bonds[4 * a + k];
    const int p0 = bond_atoms[2 * bk];
    const int p1 = bond_atoms[2 * bk + 1];
    const int other = (p0 != a) ? p0 : p1;  // select_not_equal
    eb[k] = e_n + (size_t)bk * kD;
    eo[k] = Eh  + (size_t)other * kD;
  }
  const float* fu = Fu + (size_t)atom_graph[a] * kD;
  const float* dh = Dh + (size_t)a * kD;
  float* out = h_out + (size_t)a * kD;
  for (int c = threadIdx.x; c < kD; c += blockDim.x) {
    float num = 0.0f, den = 0.0f;
#pragma unroll
    for (int k = 0; k < 4; ++k) {
      const float s = 1.0f / (1.0f + __expf(-eb[k][c]));
      num += s * eo[k][c];
      den += s;
    }
    out[c] = dh[c] + num / (den + 1e-6f) + fu[c];
  }
}

// c = 1/(|d1-d2|+1); overwrite D1<-c*d1, D2<-c*d2
__global__ void cross_gate(float* __restrict__ D1, float* __restrict__ D2,
                           size_t n) {
  size_t i = (size_t)blockIdx.x * blockDim.x + threadIdx.x;
  size_t s = (size_t)gridDim.x * blockDim.x;
  for (; i < n; i += s) {
    const float d1 = D1[i], d2 = D2[i];
    const float cc = 1.0f / (fabsf(d1 - d2) + 1.0f);
    D1[i] = cc * d1;
    D2[i] = cc * d2;
  }
}

// ============================ BatchNorm (training stats) =============
__global__ void bn_partial(const float* __restrict__ X, int rows,
                           int rowsPerBlock, float* __restrict__ ssum,
                           float* __restrict__ ssum2) {
  const int c = blockIdx.x * blockDim.x + threadIdx.x;
  if (c >= kD) return;
  const int r0 = blockIdx.y * rowsPerBlock;
  const int r1 = min(rows, r0 + rowsPerBlock);
  float s = 0.0f, s2 = 0.0f;
  for (int rr = r0; rr < r1; ++rr) {
    const float v = X[(size_t)rr * kD + c];
    s += v;
    s2 += v * v;
  }
  unsafeAtomicAdd(&ssum[c], s);
  unsafeAtomicAdd(&ssum2[c], s2);
}

__global__ void bn_finalize(const float* __restrict__ ssum,
                            const float* __restrict__ ssum2,
                            const float* __restrict__ gamma,
                            const float* __restrict__ beta, float invN,
                            float* __restrict__ sc, float* __restrict__ sh) {
  const int c = blockIdx.x * blockDim.x + threadIdx.x;
  if (c >= kD) return;
  const float m   = ssum[c] * invN;
  const float var = ssum2[c] * invN - m * m;
  const float g   = gamma[c] * rsqrtf(var + 1e-5f);
  sc[c] = g;
  sh[c] = beta[c] - m * g;
}

__global__ void bn_apply_relu(const float* __restrict__ src,
                              float* __restrict__ dst, size_t total,
                              const float* __restrict__ sc,
                              const float* __restrict__ sh) {
  size_t i = (size_t)blockIdx.x * blockDim.x + threadIdx.x;
  size_t st = (size_t)gridDim.x * blockDim.x;
  for (; i < total; i += st) {
    const int c = (int)(i % kD);
    const float v = src[i] * sc[c] + sh[c];
    dst[i] = v > 0.0f ? v : 0.0f;
  }
}

// ============================ segment counts =========================
__global__ void count_idx(const int* __restrict__ idx, int n,
                          int* __restrict__ cnt) {
  int i = blockIdx.x * blockDim.x + threadIdx.x;
  int s = gridDim.x * blockDim.x;
  for (; i < n; i += s) atomicAdd(&cnt[idx[i]], 1);
}
__global__ void inv_counts(const int* __restrict__ cnt,
                           float* __restrict__ invc, int n) {
  int i = blockIdx.x * blockDim.x + threadIdx.x;
  if (i < n) invc[i] = 1.0f / fmaxf((float)cnt[i], 1.0f);
}

// =====================================================================
extern "C" void kernel_launch(void* const* d_in, const int* in_sizes, int n_in,
                              void* d_out, int out_size, void* d_ws,
                              size_t ws_size, hipStream_t stream) {
  (void)in_sizes; (void)n_in; (void)out_size;
  const float* h1  = (const float*)d_in[0];
  const float* e1  = (const float*)d_in[1];
  const float* h2  = (const float*)d_in[2];
  const float* e2  = (const float*)d_in[3];
  const float* u   = (const float*)d_in[4];
  const float* Wall = (const float*)d_in[5];
  const float* ball = (const float*)d_in[6];
  const float* gam  = (const float*)d_in[7];
  const float* bet  = (const float*)d_in[8];
  const int* ba1 = (const int*)d_in[9];
  const int* ab1 = (const int*)d_in[10];
  const int* ag1 = (const int*)d_in[11];
  const int* bg1 = (const int*)d_in[12];
  const int* ba2 = (const int*)d_in[13];
  const int* ab2 = (const int*)d_in[14];
  const int* ag2 = (const int*)d_in[15];
  const int* bg2 = (const int*)d_in[16];

  const size_t NAD = (size_t)kNA * kD;
  const size_t NBD = (size_t)kNB * kD;
  const size_t NGD = (size_t)kNG * kD;

  // output layout: h1n | e1n | h2n | e2n | u_new
  float* out   = (float*)d_out;
  float* out_h1 = out;
  float* out_e1 = out_h1 + NAD;
  float* out_h2 = out_e1 + NBD;
  float* out_e2 = out_h2 + NAD;
  float* out_u  = out_e2 + NBD;

  // workspace carve
  const size_t need = (4 * NAD + 3 * NGD + 4 * (size_t)kD + 2048) * 4;
  if (ws_size < need) return;
  float* ws = (float*)d_ws;
  float* A    = ws;  ws += NAD;   // Ah (reused across halves)
  float* E    = ws;  ws += NAD;   // Eh (reused across halves)
  float* D1   = ws;  ws += NAD;   // Dh half 1
  float* D2   = ws;  ws += NAD;   // Dh half 2
  float* Cu   = ws;  ws += NGD;
  float* Fu   = ws;  ws += NGD;
  float* Uacc = ws;  ws += NGD;
  float* sc   = ws;  ws += kD;
  float* sh   = ws;  ws += kD;
  float* ssum = ws;  ws += 2 * kD;  // ssum | ssum2
  float* invc = ws;  ws += 1024;
  int*   cnt  = (int*)ws;           // 1024 ints

  auto Wp = [&](int i) { return Wall + (size_t)i * kD * kD; };
  auto bp = [&](int i) { return ball + (size_t)i * kD; };
  auto gblocks = [&](int M) { return ((M / 32) * 10 + 3) / 4; };

  auto gemm_plain = [&](const float* X, int wi, float* Y, int M) {
    gemm800<0><<<gblocks(M), 128, 0, stream>>>(X, Wp(wi), bp(wi), Y, M, 0.f,
                                               0.f, nullptr, nullptr, nullptr);
  };
  auto gemm_blend = [&](const float* X, int wi, float* Y, int M, float al,
                        float be) {
    gemm800<1><<<gblocks(M), 128, 0, stream>>>(X, Wp(wi), bp(wi), Y, M, al, be,
                                               nullptr, nullptr, nullptr);
  };
  auto gemm_seg = [&](const float* X, int wi, const int* seg, const float* ic,
                      int M) {
    gemm800<2><<<gblocks(M), 128, 0, stream>>>(X, Wp(wi), bp(wi), nullptr, M,
                                               0.f, 0.f, seg, ic, Uacc);
  };
  auto bn = [&](const float* X, float* dst, int rows, int bnidx) {
    zero_f32<<<2, 256, 0, stream>>>(ssum, 2 * (size_t)kD);
    const int rpb = 512;
    const int ch = (rows + rpb - 1) / rpb;
    bn_partial<<<dim3(4, ch), 256, 0, stream>>>(X, rows, rpb, ssum, ssum + kD);
    bn_finalize<<<4, 256, 0, stream>>>(ssum, ssum + kD, gam + (size_t)bnidx * kD,
                                       bet + (size_t)bnidx * kD,
                                       1.0f / (float)rows, sc, sh);
    bn_apply_relu<<<2048, 256, 0, stream>>>(X, dst, (size_t)rows * kD, sc, sh);
  };

  // ---- shared global-node projections (u is the same for both halves)
  gemm_plain(u, 2, Cu, kNG);  // Cu = u@C+c
  gemm_plain(u, 5, Fu, kNG);  // Fu = u@F+f

  // ---- half processing
  auto run_half = [&](const float* h, const float* e, const int* ba,
                      const int* ab, const int* ag, const int* bg, float* Dh,
                      float* out_h, float* out_e) {
    gemm_plain(h, 0, A, kNA);       // Ah
    gemm_plain(h, 4, E, kNA);       // Eh
    gemm_plain(h, 3, Dh, kNA);      // Dh
    gemm_plain(e, 1, out_e, kNB);   // Be -> e out region
    bond_gather<<<kNB, 256, 0, stream>>>(out_e, A, Cu, ba, bg);
    bn(out_e, out_e, kNB, 0);       // e_new = relu(bn(...)), in place
    atom_update<<<kNA, 256, 0, stream>>>(out_h, Dh, E, Fu, out_e, ab, ba, ag);
  };
  run_half(h1, e1, ba1, ab1, ag1, bg1, D1, out_h1, out_e1);
  run_half(h2, e2, ba2, ab2, ag2, bg2, D2, out_h2, out_e2);

  // ---- cross u2v/v2u: D1<-c*Dh1, D2<-c*Dh2, then blend through W9
  cross_gate<<<2048, 256, 0, stream>>>(D1, D2, NAD);
  gemm_blend(D2, 9, out_h1, kNA, 0.2f, 0.8f);  // h1n = 0.8*h1n + 0.2*m21
  gemm_blend(D1, 9, out_h2, kNA, 0.2f, 0.8f);  // h2n = 0.8*h2n + 0.2*m12
  bn(out_h1, out_h1, kNA, 1);
  bn(out_h2, out_h2, kNA, 1);

  // ---- global update: fused segment means + Iu, then BN+relu
  zero_i32<<<4, 256, 0, stream>>>(cnt, 1024);
  zero_f32<<<256, 256, 0, stream>>>(Uacc, NGD);
  count_idx<<<40, 256, 0, stream>>>(ag1, kNA, cnt);
  count_idx<<<48, 256, 0, stream>>>(bg1, kNB, cnt + 256);
  count_idx<<<40, 256, 0, stream>>>(ag2, kNA, cnt + 512);
  count_idx<<<48, 256, 0, stream>>>(bg2, kNB, cnt + 768);
  inv_counts<<<4, 256, 0, stream>>>(cnt, invc, 1024);
  gemm_seg(out_h1, 6, ag1, invc,       kNA);  // mean(Gh1)
  gemm_seg(out_e1, 7, bg1, invc + 256, kNB);  // mean(He1)
  gemm_seg(out_h2, 6, ag2, invc + 512, kNA);  // mean(Gh2)
  gemm_seg(out_e2, 7, bg2, invc + 768, kNB);  // mean(He2)
  gemm_blend(u, 8, Uacc, kNG, 1.0f, 1.0f);    // += Iu
  bn(Uacc, out_u, kNG, 2);
}